// SelfAttention2d_64879775973586
// MI455X (gfx1250) — compile-verified
//
#include <hip/hip_runtime.h>
#include <math.h>

typedef __attribute__((ext_vector_type(2))) float v2f;
typedef __attribute__((ext_vector_type(8))) float v8f;

#define CIN   128
#define COUT  128
#define OC3   384      // 3*COUT
#define NG    8        // groups
#define GC    16       // channels per group
#define HH    64
#define WW    64
#define HWP   4096     // H*W
#define KW    7
#define PADR  3
#define KK    49

// ---------------------------------------------------------------------------
// Kernel 1: qkv[n][o][p] = sum_c w[o][c] * x[n][c][p]
// WMMA f32 16x16x4, one 16(M)x16(cols) tile per wave, K-loop over 128.
// ---------------------------------------------------------------------------
__global__ __launch_bounds__(256) void qkv_gemm_wmma(
    const float* __restrict__ xg,    // (N,128,4096)
    const float* __restrict__ wq,    // (384,128)
    float* __restrict__ qkv,         // (N,384,4096) workspace
    int numTiles)
{
  const int wave = threadIdx.x >> 5;
  const int lane = threadIdx.x & 31;
  const int t = blockIdx.x * 8 + wave;
  if (t >= numTiles) return;               // wave-uniform; EXEC stays all-1s

  const int mTile   = t % 24;              // 384/16 output-channel tiles
  const int colTile = t / 24;              // pixel-column tiles
  const int mBase   = mTile * 16;
  const int col     = colTile * 16 + (lane & 15);   // global column (n*4096+p)
  const int n       = col >> 12;
  const int p       = col & 4095;
  const int hi      = lane >> 4;

  // A-frag: row m = mBase + (lane&15); k = k0 + 2*hi + {0,1}  (contiguous pair)
  const float* wrow = wq + (size_t)(mBase + (lane & 15)) * CIN + 2 * hi;
  // B-frag: col fixed per lane; k = k0 + hi + {0,2}
  const float* xcol = xg + (size_t)n * (CIN * HWP) + (size_t)hi * HWP + p;

  v8f acc = {0.f, 0.f, 0.f, 0.f, 0.f, 0.f, 0.f, 0.f};
  #pragma unroll 8
  for (int k0 = 0; k0 < CIN; k0 += 4) {
    v2f a = *(const v2f*)(wrow + k0);
    v2f b;
    b.x = xcol[(size_t)k0 * HWP];
    b.y = xcol[(size_t)(k0 + 2) * HWP];
    acc = __builtin_amdgcn_wmma_f32_16x16x4_f32(
        /*neg_a=*/false, a, /*neg_b=*/false, b,
        /*c_mod=*/(short)0, acc, /*reuse_a=*/false, /*reuse_b=*/false);
  }

  // C/D layout: VGPR r -> M = r (lanes 0-15) / r+8 (lanes 16-31), N = lane&15
  float* outb = qkv + (size_t)n * OC3 * HWP + p;
  #pragma unroll
  for (int r = 0; r < 8; ++r) {
    int m = mBase + r + 8 * hi;
    outb[(size_t)m * HWP] = acc[r];
  }
}

// ---------------------------------------------------------------------------
// Kernel 2: local 7x7 attention per (n, g, pixel). 16x16 pixel tile per block,
// k/v halos (22x22x16ch) staged in LDS, softmax over 49, weighted V sum.
// ---------------------------------------------------------------------------
#define TILE  16
#define HALO  22          // TILE + 2*PADR
#define PITCH 23          // padded row to dodge bank conflicts

__global__ __launch_bounds__(256) void attn_local7x7(
    const float* __restrict__ qkv,   // (N,384,4096)
    const float* __restrict__ relh,  // (8,7)
    const float* __restrict__ relw,  // (8,7)
    float* __restrict__ out)         // (N,128,4096)
{
  __shared__ float kT[GC * HALO * PITCH];
  __shared__ float vT[GC * HALO * PITCH];

  const int blk    = blockIdx.x;           // n*128 + g*16 + tile
  const int tileId = blk & 15;
  const int g      = (blk >> 4) & 7;
  const int n      = blk >> 7;
  const int ty0    = (tileId >> 2) * TILE;
  const int tx0    = (tileId & 3) * TILE;

  const float* kbase = qkv + ((size_t)n * OC3 + COUT + g * GC) * HWP;
  const float* vbase = kbase + (size_t)COUT * HWP;

  // Cooperative halo load (zero-pad out of bounds == unfold's zero padding)
  for (int e = threadIdx.x; e < GC * HALO * HALO; e += 256) {
    int c   = e / (HALO * HALO);
    int rem = e - c * (HALO * HALO);
    int r   = rem / HALO;
    int q   = rem - r * HALO;
    int y   = ty0 - PADR + r;
    int x   = tx0 - PADR + q;
    float kv_ = 0.f, vv_ = 0.f;
    if ((unsigned)y < HH && (unsigned)x < WW) {
      size_t off = (size_t)c * HWP + (size_t)y * WW + x;
      kv_ = kbase[off];
      vv_ = vbase[off];
    }
    kT[c * (HALO * PITCH) + r * PITCH + q] = kv_;
    vT[c * (HALO * PITCH) + r * PITCH + q] = vv_;
  }
  __syncthreads();

  const int tx = threadIdx.x & 15;
  const int ty = threadIdx.x >> 4;
  const size_t pix = (size_t)(ty0 + ty) * WW + (tx0 + tx);

  // Load q (16 channels for this pixel/group)
  const float* qbase = qkv + ((size_t)n * OC3 + g * GC) * HWP + pix;
  float qv[GC];
  #pragma unroll
  for (int c = 0; c < GC; ++c) qv[c] = qbase[(size_t)c * HWP];

  // Separable relative bias: bias(dy,dx) = biasH[dy] + biasW[dx]
  float biasH[KW], biasW[KW];
  #pragma unroll
  for (int d = 0; d < KW; ++d) {
    float sh = 0.f, sw = 0.f;
    #pragma unroll
    for (int c = 0; c < GC / 2; ++c) {
      sh += qv[c]          * relh[c * KW + d];
      sw += qv[GC / 2 + c] * relw[c * KW + d];
    }
    biasH[d] = sh;
    biasW[d] = sw;
  }

  // Logits + running max
  float logit[KK];
  float mx = -3.402823466e+38f;
  #pragma unroll
  for (int dy = 0; dy < KW; ++dy) {
    #pragma unroll
    for (int dx = 0; dx < KW; ++dx) {
      float s = biasH[dy] + biasW[dx];
      int base = (ty + dy) * PITCH + (tx + dx);
      #pragma unroll
      for (int c = 0; c < GC; ++c)
        s += qv[c] * kT[c * (HALO * PITCH) + base];
      logit[dy * KW + dx] = s;
      mx = fmaxf(mx, s);
    }
  }

  // Softmax
  float denom = 0.f;
  #pragma unroll
  for (int i = 0; i < KK; ++i) {
    float e = __expf(logit[i] - mx);
    logit[i] = e;
    denom += e;
  }
  const float inv = 1.f / denom;

  // Weighted V sum
  float acc[GC];
  #pragma unroll
  for (int c = 0; c < GC; ++c) acc[c] = 0.f;
  #pragma unroll
  for (int dy = 0; dy < KW; ++dy) {
    #pragma unroll
    for (int dx = 0; dx < KW; ++dx) {
      float a = logit[dy * KW + dx];
      int base = (ty + dy) * PITCH + (tx + dx);
      #pragma unroll
      for (int c = 0; c < GC; ++c)
        acc[c] += a * vT[c * (HALO * PITCH) + base];
    }
  }

  float* ob = out + ((size_t)n * COUT + g * GC) * HWP + pix;
  #pragma unroll
  for (int c = 0; c < GC; ++c) ob[(size_t)c * HWP] = acc[c] * inv;
}

// ---------------------------------------------------------------------------
extern "C" void kernel_launch(void* const* d_in, const int* in_sizes, int n_in,
                              void* d_out, int out_size, void* d_ws, size_t ws_size,
                              hipStream_t stream)
{
  const float* x  = (const float*)d_in[0];   // (N,128,64,64)
  const float* wq = (const float*)d_in[1];   // (384,128)
  const float* rh = (const float*)d_in[2];   // (8,1,1,7,1) -> flat (8,7)
  const float* rw = (const float*)d_in[3];   // (8,1,1,1,7) -> flat (8,7)
  float* out = (float*)d_out;
  float* qkv = (float*)d_ws;                 // N*384*4096 floats (~25 MB for N=4)

  const int N = in_sizes[0] / (CIN * HWP);

  // Kernel 1: 24 M-tiles x (N*4096/16) column tiles, 8 waves (tiles) per block
  const int numTiles = 24 * (N * HWP / 16);
  const int blocks1  = (numTiles + 7) / 8;
  qkv_gemm_wmma<<<dim3(blocks1), dim3(256), 0, stream>>>(x, wq, qkv, numTiles);

  // Kernel 2: one block per (n, g, 16x16 pixel tile)
  const int blocks2 = N * NG * 16;
  attn_local7x7<<<dim3(blocks2), dim3(256), 0, stream>>>(qkv, rh, rw, out);
}